// BoundsChecker_42099269435917
// MI455X (gfx1250) — compile-verified
//
#include <hip/hip_runtime.h>
#include <hip/hip_bf16.h>

typedef __attribute__((ext_vector_type(2))) float v2f;
typedef __attribute__((ext_vector_type(8))) float v8f;

#define CHUNK_TILES 128            // 128 tiles * 32 entries * 8B = 32KB LDS per chunk
#define POS_PER_BLOCK 128          // 8 waves * 16 rows
#define BIGF 3.402823466e38f

// ---------------------------------------------------------------------------
// Kernel 1: WMMA-based argmin over the [B,N] squared-distance matrix.
//   d2_rel[m][n] = |q_n|^2 - 2 p_m . q_n   (argmin-equivalent to full d2)
// computed as a 16x16x4 f32 WMMA:  A row = (-2px,-2py,1,0), B col = (qx,qy,|q|^2,0)
// ---------------------------------------------------------------------------
__global__ void __launch_bounds__(256)
bc_argmin_kernel(const float* __restrict__ pos,
                 const float* __restrict__ refl,
                 float* __restrict__ r0_out,
                 int B, int N) {
    __shared__ v2f  ldsB[CHUNK_TILES * 32];    // B-operand, pre-formatted per lane
    __shared__ float redV[POS_PER_BLOCK][16];
    __shared__ int   redI[POS_PER_BLOCK][16];

    const int tid  = threadIdx.x;
    const int lane = tid & 31;
    const int wave = tid >> 5;
    const int n16  = lane & 15;
    const int halfsel = lane >> 4;             // 0: lanes 0-15, 1: lanes 16-31

    // ---- A matrix (16x4 f32): lanes 0-15 hold K=0,1 ; lanes 16-31 hold K=2,3
    const int posBase = blockIdx.x * POS_PER_BLOCK + wave * 16;
    int pidx = posBase + n16; if (pidx > B - 1) pidx = B - 1;
    const float px = pos[2 * pidx];
    const float py = pos[2 * pidx + 1];
    v2f amat;
    amat.x = halfsel ? 1.0f : (-2.0f * px);    // K=0 or K=2
    amat.y = halfsel ? 0.0f : (-2.0f * py);    // K=1 or K=3

    // ---- running minima: slot i holds row m = i + 8*halfsel, column n16
    float bestV[8];
    int   bestT[8];
#pragma unroll
    for (int i = 0; i < 8; ++i) { bestV[i] = BIGF; bestT[i] = 0; }

    const int numTiles  = N >> 4;                                   // N/16
    const int numChunks = (numTiles + CHUNK_TILES - 1) / CHUNK_TILES;

    for (int c = 0; c < numChunks; ++c) {
        const int tileBase = c * CHUNK_TILES;
        int tilesInChunk = numTiles - tileBase;
        if (tilesInChunk > CHUNK_TILES) tilesInChunk = CHUNK_TILES;

        // ---- cooperative fill of B-operand into LDS (wave-uniform loop)
        for (int e = tid; e < tilesInChunk * 32; e += 256) {
            const int tileL = e >> 5;
            const int laneE = e & 31;
            int n = (tileBase + tileL) * 16 + (laneE & 15);
            if (n > N - 1) n = N - 1;
            const float qx = refl[2 * n];
            const float qy = refl[2 * n + 1];
            v2f bv;
            if (laneE < 16) { bv.x = qx; bv.y = qy; }                 // K=0,1
            else            { bv.x = qx * qx + qy * qy; bv.y = 0.0f; } // K=2,3
            ldsB[e] = bv;
        }
        __syncthreads();

        // ---- WMMA sweep over tiles (EXEC all-ones, no divergence)
#pragma unroll 4
        for (int tl = 0; tl < tilesInChunk; ++tl) {
            v2f bmat = ldsB[tl * 32 + lane];
            v8f acc = {};
            acc = __builtin_amdgcn_wmma_f32_16x16x4_f32(
                false, amat, false, bmat, (short)0, acc, false, false);
            const int t = tileBase + tl;
#pragma unroll
            for (int i = 0; i < 8; ++i) {
                const float di = acc[i];
                const bool lt = di < bestV[i];
                bestV[i] = lt ? di : bestV[i];
                bestT[i] = lt ? t  : bestT[i];
            }
        }
        __syncthreads();
    }

    // ---- cross-lane reduction through LDS: per row m, min over 16 lane-columns
#pragma unroll
    for (int i = 0; i < 8; ++i) {
        const int m = i + 8 * halfsel;
        const int row = wave * 16 + m;
        redV[row][n16] = bestV[i];
        redI[row][n16] = bestT[i] * 16 + n16;   // global refline index
    }
    __syncthreads();

    if (tid < POS_PER_BLOCK) {
        const int b = blockIdx.x * POS_PER_BLOCK + tid;
        if (b < B) {
            float bv = redV[tid][0];
            int   bi = redI[tid][0];
#pragma unroll
            for (int j = 1; j < 16; ++j) {
                const float v = redV[tid][j];
                if (v < bv) { bv = v; bi = redI[tid][j]; }
            }
            r0_out[b] = (float)bi;              // DR_SAMP == 1
        }
    }
}

// ---------------------------------------------------------------------------
// Kernel 2: masked Newton refinement + final path eval / width interp.
// ---------------------------------------------------------------------------
__global__ void __launch_bounds__(256)
bc_newton_kernel(const float* __restrict__ pos,
                 const float* __restrict__ refl,
                 const float* __restrict__ lw,
                 const float* __restrict__ rw,
                 const int*   __restrict__ nit,
                 const float* __restrict__ r0,
                 float* __restrict__ out,
                 int B, int N) {
    const int b = blockIdx.x * blockDim.x + threadIdx.x;
    if (b >= B) return;

    const float L = (float)N;
    const float px = pos[2 * b];
    const float py = pos[2 * b + 1];
    float r = r0[b];
    const int iters = nit[0];
    bool done = false;

    for (int it = 0; it < iters; ++it) {
        float u = fmodf(r, L); if (u < 0.0f) u += L;
        const float fu = floorf(u);
        int i0 = (int)fu; i0 %= N; if (i0 < 0) i0 += N;
        int i1 = i0 + 1; if (i1 >= N) i1 = 0;
        const float frac = u - fu;
        const float p0x = refl[2 * i0], p0y = refl[2 * i0 + 1];
        const float p1x = refl[2 * i1], p1y = refl[2 * i1 + 1];
        const float dxs = p1x - p0x, dys = p1y - p0y;
        const float ptx = p0x + frac * dxs;
        const float pty = p0y + frac * dys;
        const float invlen = 1.0f / sqrtf(dxs * dxs + dys * dys);
        const float tx = dxs * invlen, ty = dys * invlen;
        const float ex = px - ptx, ey = py - pty;
        const float fprime = -(ex * tx + ey * ty);
        float step = -fprime;
        step = fminf(1.0f, fmaxf(-1.0f, step));         // MAX_STEP = 1
        const float add = done ? 0.0f : step;           // NEWTON_STEPSIZE = 1
        r = fmodf(r + add, L); if (r < 0.0f) r += L;
        done = done || (fabsf(fprime) < 1e-4f) || (fabsf(step) < 1e-2f);
    }

    // final path eval at r
    float u = fmodf(r, L); if (u < 0.0f) u += L;
    const float fu = floorf(u);
    int i0 = (int)fu; i0 %= N; if (i0 < 0) i0 += N;
    int i1 = i0 + 1; if (i1 >= N) i1 = 0;
    const float frac = u - fu;
    const float p0x = refl[2 * i0], p0y = refl[2 * i0 + 1];
    const float p1x = refl[2 * i1], p1y = refl[2 * i1 + 1];
    const float dxs = p1x - p0x, dys = p1y - p0y;
    const float ptx = p0x + frac * dxs;
    const float pty = p0y + frac * dys;
    const float invlen = 1.0f / sqrtf(dxs * dxs + dys * dys);
    const float tx = dxs * invlen, ty = dys * invlen;
    const float nx = -ty, ny = tx;                      // 90 deg CCW
    const float ex = px - ptx, ey = py - pty;
    const float nproj = ex * nx + ey * ny;
    const float lwv = lw[i0] * (1.0f - frac) + lw[i1] * frac;
    const float rwv = rw[i0] * (1.0f - frac) + rw[i1] * frac;

    // flat concat: r[B], point[B,2], tang[B,2], norm[B,2], deltas[B,2],
    //              nproj[B], lw[B], rw[B]
    out[b] = r;
    out[B + 2 * b]         = ptx;  out[B + 2 * b + 1]     = pty;
    out[3 * B + 2 * b]     = tx;   out[3 * B + 2 * b + 1] = ty;
    out[5 * B + 2 * b]     = nx;   out[5 * B + 2 * b + 1] = ny;
    out[7 * B + 2 * b]     = ex;   out[7 * B + 2 * b + 1] = ey;
    out[9 * B + b]  = nproj;
    out[10 * B + b] = lwv;
    out[11 * B + b] = rwv;
}

extern "C" void kernel_launch(void* const* d_in, const int* in_sizes, int n_in,
                              void* d_out, int out_size, void* d_ws, size_t ws_size,
                              hipStream_t stream) {
    const float* pos  = (const float*)d_in[0];
    const float* refl = (const float*)d_in[1];
    const float* lw   = (const float*)d_in[2];
    const float* rw   = (const float*)d_in[3];
    const int*   nit  = (const int*)d_in[4];
    const int B = in_sizes[0] / 2;
    const int N = in_sizes[1] / 2;

    float* r0  = (float*)d_ws;       // B floats of scratch
    float* out = (float*)d_out;

    const int blocks1 = (B + POS_PER_BLOCK - 1) / POS_PER_BLOCK;
    bc_argmin_kernel<<<blocks1, 256, 0, stream>>>(pos, refl, r0, B, N);

    const int blocks2 = (B + 255) / 256;
    bc_newton_kernel<<<blocks2, 256, 0, stream>>>(pos, refl, lw, rw, nit, r0, out, B, N);
}